// Causal_TransProb_31971736551502
// MI455X (gfx1250) — compile-verified
//
#include <hip/hip_runtime.h>
#include <math.h>

// Problem constants
#define Bsz 16
#define Tt  12
#define NM  512
#define NN  512
#define Hh  256
#define H4  64
#define Rr  3
#define KE  768          // e-part of embed K (H4*T); z-part is the other 768
#define NEGV (-1.0e9f)
#define DROPV 0.6f

typedef float f32x2 __attribute__((ext_vector_type(2)));
typedef float f32x8 __attribute__((ext_vector_type(8)));

// ---------------- workspace layout (float offsets) ----------------
// 0         : z_date   (12288)
// 12288     : z_weather(12288)
// 24576     : zc       (16384)
// 40960     : ex[4]    (8388608)   ends 8429568
// 8429568   : Xm_e     (6291456)
// 14721024  : Xn_e     (6291456)   ends 21012480
// 8429568   : qk[4]    (8388608)   -- reuses X region after embeds
// 16818176  : sim[4]   (16777216)  ends 33595392 (~134.4 MB)
#define OFF_ZD  0
#define OFF_ZW  12288
#define OFF_ZC  24576
#define OFF_EX  40960
#define OFF_XM  8429568
#define OFF_XN  14721024
#define OFF_QK  8429568
#define OFF_SIM 16818176
#define EX_SZ   2097152
#define QK_SZ   2097152
#define SIM_SZ  4194304

// ---------------- kernel 1: time/weather convs ----------------
__global__ void conv_z_kernel(const float* __restrict__ tx, const float* __restrict__ twt,
                              const float* __restrict__ tb, const float* __restrict__ wx,
                              const float* __restrict__ wwt, const float* __restrict__ wb,
                              float* __restrict__ zd, float* __restrict__ zw) {
    int i = blockIdx.x * blockDim.x + threadIdx.x;
    if (i >= Bsz * H4 * Tt) return;
    int t = i % Tt, o = (i / Tt) % H4, b = i / (Tt * H4);
    float accd = tb[o];
    for (int c = 0; c < 2; ++c)
        for (int k = 0; k < 7; ++k)
            accd += tx[((b * (Tt + 6)) + (t + k)) * 2 + c] * twt[(o * 2 + c) * 7 + k];
    zd[i] = accd;
    float accw = wb[o];
    for (int c = 0; c < 2; ++c)
        for (int k = 0; k < 5; ++k)
            accw += wx[((b * (Tt + 4)) + (t + k)) * 2 + c] * wwt[(o * 2 + c) * 5 + k];
    zw[i] = accw;
}

// ---------------- kernel 2: build node-feature embed inputs ----------------
// Xm[(b*512+n)*768 + c*12+t] = xm[b,t,n,:] . conv_xm_w[c,:] + conv_xm_b[c]
__global__ void build_x_kernel(const float* __restrict__ xm, const float* __restrict__ xn,
                               const float* __restrict__ wm, const float* __restrict__ bm,
                               const float* __restrict__ wn, const float* __restrict__ bn,
                               float* __restrict__ Xm, float* __restrict__ Xn) {
    int i = blockIdx.x * blockDim.x + threadIdx.x;
    if (i >= Bsz * NM * KE) return;
    int k = i % KE, n = (i / KE) % NM, b = i / (KE * NM);
    int c = k / Tt, t = k % Tt;
    const float* px = xm + (((b * Tt + t) * NM) + n) * 2;
    Xm[i] = px[0] * wm[c * 2] + px[1] * wm[c * 2 + 1] + bm[c];
    const float* qx = xn + (((b * Tt + t) * NN) + n) * 2;
    Xn[i] = qx[0] * wn[c * 2] + qx[1] * wn[c * 2 + 1] + bn[c];
}

// ---------------- kernel 3: z part of embed (per batch, node independent) ----------------
// zc[g, b, h] = sum_k z_sel[b, k] * W_g[h, 768 + k]
__global__ void zcontrib_kernel(const float* __restrict__ zd, const float* __restrict__ zw,
                                const float* __restrict__ w0, const float* __restrict__ w1,
                                const float* __restrict__ w2, const float* __restrict__ w3,
                                float* __restrict__ zc) {
    int i = blockIdx.x * blockDim.x + threadIdx.x;
    if (i >= 4 * Bsz * Hh) return;
    int h = i & 255, b = (i >> 8) & 15, g = i >> 12;
    const float* W = (g == 0) ? w0 : (g == 1) ? w1 : (g == 2) ? w2 : w3;
    const float* z = (g & 1) ? zw : zd;
    const float* wr = W + h * (2 * KE) + KE;
    const float* zr = z + b * KE;
    float acc = 0.f;
    for (int k = 0; k < KE; ++k) acc += zr[k] * wr[k];
    zc[i] = acc;
}

// ---------------- fp32 WMMA, 16x64 tile per wave (4x register blocking along N) ----------------
// A lane layout (ISA 7.12.2): lane<16 -> M=lane, K={kb,kb+1}; lane>=16 -> M=lane-16, K={kb+2,kb+3}.
// B mirrors the K-half split. C/D: VGPR v -> M = v + 8*(lane>=16), N = lane&15.
// One A-fragment load feeds 4 WMMAs -> load:wmma = 1.25:1 instead of 2:1.
__device__ __forceinline__ void wmma_f32_tile4(const float* __restrict__ A, int lda,
                                               const float* __restrict__ Bm, int ldb,
                                               int K, int lane, f32x8 acc[4]) {
    int row = lane & 15;
    int kh = (lane >> 4) << 1;  // 0 or 2
    const float* ap = A + row * lda + kh;
    const float* bp = Bm + row * ldb + kh;
    const int ns = 16 * ldb;    // stride between consecutive 16-wide N tiles
#pragma unroll 2
    for (int k = 0; k < K; k += 4) {
        f32x2 a, b0, b1, b2, b3;
        a[0] = ap[0];           a[1] = ap[1];
        b0[0] = bp[0];          b0[1] = bp[1];
        b1[0] = bp[ns];         b1[1] = bp[ns + 1];
        b2[0] = bp[2 * ns];     b2[1] = bp[2 * ns + 1];
        b3[0] = bp[3 * ns];     b3[1] = bp[3 * ns + 1];
        acc[0] = __builtin_amdgcn_wmma_f32_16x16x4_f32(false, a, false, b0, (short)0, acc[0],
                                                       false, false);
        acc[1] = __builtin_amdgcn_wmma_f32_16x16x4_f32(false, a, false, b1, (short)0, acc[1],
                                                       false, false);
        acc[2] = __builtin_amdgcn_wmma_f32_16x16x4_f32(false, a, false, b2, (short)0, acc[2],
                                                       false, false);
        acc[3] = __builtin_amdgcn_wmma_f32_16x16x4_f32(false, a, false, b3, (short)0, acc[3],
                                                       false, false);
        ap += 4; bp += 4;
    }
}

// ---------------- kernel 4: embed GEMM (WMMA) ----------------
// ex[g][(b*512+n), h] = X[(b*512+n), :768] @ W_g[h, :768] + zc[g,b,h] + bias_g[h]
__global__ void embed_gemm_kernel(const float* __restrict__ Xm, const float* __restrict__ Xn,
                                  const float* __restrict__ w0, const float* __restrict__ w1,
                                  const float* __restrict__ w2, const float* __restrict__ w3,
                                  const float* __restrict__ b0, const float* __restrict__ b1,
                                  const float* __restrict__ b2, const float* __restrict__ b3,
                                  const float* __restrict__ zc, float* __restrict__ ex) {
    int g = blockIdx.z;
    const float* X = (g < 2) ? Xm : Xn;
    const float* W = (g == 0) ? w0 : (g == 1) ? w1 : (g == 2) ? w2 : w3;
    const float* bias = (g == 0) ? b0 : (g == 1) ? b1 : (g == 2) ? b2 : b3;
    int lane = threadIdx.x & 31;
    int wv = threadIdx.x >> 5;
    int m0 = (blockIdx.x * 8 + wv) * 16;
    int n0 = blockIdx.y * 64;
    f32x8 acc[4] = {{0.f}, {0.f}, {0.f}, {0.f}};
    wmma_f32_tile4(X + m0 * KE, KE, W + n0 * (2 * KE), 2 * KE, KE, lane, acc);
    int rs = (lane >> 4) * 8;
    int col = lane & 15;
    int b = m0 >> 9;
#pragma unroll
    for (int nb = 0; nb < 4; ++nb) {
        int n = n0 + nb * 16 + col;
        float add = zc[(g << 12) + (b << 8) + n] + bias[n];
        float* C = ex + g * EX_SZ + m0 * Hh + n;
#pragma unroll
        for (int v = 0; v < 8; ++v) C[(v + rs) * Hh] = acc[nb][v] + add;
    }
}

// ---------------- kernel 5: qk GEMM (WMMA): qk[g] = ex[g&1] @ P_g^T ----------------
__global__ void qk_gemm_kernel(const float* __restrict__ ex,
                               const float* __restrict__ p0, const float* __restrict__ p1,
                               const float* __restrict__ p2, const float* __restrict__ p3,
                               float* __restrict__ qk) {
    int g = blockIdx.z;
    const float* A = ex + (g & 1) * EX_SZ;
    const float* P = (g == 0) ? p0 : (g == 1) ? p1 : (g == 2) ? p2 : p3;
    int lane = threadIdx.x & 31;
    int wv = threadIdx.x >> 5;
    int m0 = (blockIdx.x * 8 + wv) * 16;
    int n0 = blockIdx.y * 64;
    f32x8 acc[4] = {{0.f}, {0.f}, {0.f}, {0.f}};
    wmma_f32_tile4(A + m0 * Hh, Hh, P + n0 * Hh, Hh, Hh, lane, acc);
    int rs = (lane >> 4) * 8;
    int col = lane & 15;
#pragma unroll
    for (int nb = 0; nb < 4; ++nb) {
        float* C = qk + g * QK_SZ + m0 * Hh + n0 + nb * 16 + col;
#pragma unroll
        for (int v = 0; v < 8; ++v) C[(v + rs) * Hh] = acc[nb][v];
    }
}

// ---------------- kernel 6: sim GEMM (WMMA, batched): sim[g][b] = qk[g][b] @ v_g[b]^T ----------------
__global__ void sim_gemm_kernel(const float* __restrict__ qk, const float* __restrict__ ex,
                                float* __restrict__ sim) {
    int g = blockIdx.z;
    int b = blockIdx.y;
    int wv = threadIdx.x >> 5;
    int lane = threadIdx.x & 31;
    int tile = blockIdx.x * 8 + wv;        // 0..255 : 32 m-tiles x 8 n-blocks
    int mt = tile >> 3, ntb = tile & 7;    // waves in a block share mt -> shared A fragments
    int n0 = ntb * 64;
    const float* A = qk + g * QK_SZ + (b * NM + mt * 16) * Hh;
    const float* V = ex + (2 + (g & 1)) * EX_SZ + (b * NN + n0) * Hh;
    f32x8 acc[4] = {{0.f}, {0.f}, {0.f}, {0.f}};
    wmma_f32_tile4(A, Hh, V, Hh, Hh, lane, acc);
    int rs = (lane >> 4) * 8;
    int col = lane & 15;
#pragma unroll
    for (int nb = 0; nb < 4; ++nb) {
        float* C = sim + g * SIM_SZ + b * (NM * NN) + (mt * 16) * NN + n0 + nb * 16 + col;
#pragma unroll
        for (int v = 0; v < 8; ++v) C[(v + rs) * NN] = acc[nb][v];
    }
}

// ---------------- wave32 reductions ----------------
__device__ __forceinline__ float wave_max32(float v) {
#pragma unroll
    for (int o = 16; o > 0; o >>= 1) v = fmaxf(v, __shfl_xor(v, o, 32));
    return v;
}
__device__ __forceinline__ float wave_sum32(float v) {
#pragma unroll
    for (int o = 16; o > 0; o >>= 1) v += __shfl_xor(v, o, 32);
    return v;
}

// ---------------- kernel 7: m2n masked softmax (axis = n, contiguous) ----------------
__global__ void m2n_softmax_kernel(const float* __restrict__ sim, const float* __restrict__ adj,
                                   float* __restrict__ out) {
    int lane = threadIdx.x & 31;
    int rowid = blockIdx.x * 8 + (threadIdx.x >> 5);   // b*512+m, 0..8191
    int b = rowid >> 9, m = rowid & 511;
    const float* pd = sim + 0 * SIM_SZ + rowid * NN;
    const float* pw = sim + 1 * SIM_SZ + rowid * NN;
    float sd[16], sw[16], ed[16], ew[16];
#pragma unroll
    for (int j = 0; j < 16; ++j) { sd[j] = pd[lane + 32 * j]; sw[j] = pw[lane + 32 * j]; }
    for (int r = 0; r < Rr; ++r) {
        const float* gr = adj + (r * NM + m) * NN;
        float mxd = -3.402823466e38f, mxw = -3.402823466e38f;
#pragma unroll
        for (int j = 0; j < 16; ++j) {
            float g = gr[lane + 32 * j];
            float x = sd[j]; if (g == 0.f) x = 0.f; if (x == 0.f) x = NEGV;
            sd[j] = x; mxd = fmaxf(mxd, x);
            float y = sw[j]; if (g == 0.f) y = 0.f; if (y == 0.f) y = NEGV;
            sw[j] = y; mxw = fmaxf(mxw, y);
        }
        mxd = wave_max32(mxd); mxw = wave_max32(mxw);
        float smd = 0.f, smw = 0.f;
#pragma unroll
        for (int j = 0; j < 16; ++j) {
            ed[j] = __expf(sd[j] - mxd); smd += ed[j];
            ew[j] = __expf(sw[j] - mxw); smw += ew[j];
        }
        smd = wave_sum32(smd); smw = wave_sum32(smw);
        float rd = 1.f / smd, rw = 1.f / smw;
        float* po = out + ((b * Rr + r) * NM + m) * NN;
#pragma unroll
        for (int j = 0; j < 16; ++j) {
            float p1 = ed[j] * rd; p1 = (p1 >= DROPV) ? p1 : 0.f;
            float p2 = ew[j] * rw; p2 = (p2 >= DROPV) ? p2 : 0.f;
            po[lane + 32 * j] = 0.5f * (p1 + p2);
        }
    }
}

// ---------------- kernel 8: n2m masked softmax (axis = m, strided) via LDS tiles ----------------
#define NCH 16      // n-columns per block
#define PAD 17      // padded LDS row stride (floats) to dodge 64-bank conflicts
__global__ void n2m_softmax_kernel(const float* __restrict__ sim, const float* __restrict__ adj,
                                   float* __restrict__ out) {
    extern __shared__ float smem[];
    float* td = smem;                    // 512*17
    float* tw = smem + NM * PAD;         // 512*17
    float* red = smem + 2 * NM * PAD;    // 256
    int tid = threadIdx.x;
    int n0 = blockIdx.x * NCH;
    int b = blockIdx.y;
    const float* sd = sim + 2 * SIM_SZ + b * (NM * NN);
    const float* sw = sim + 3 * SIM_SZ + b * (NM * NN);
    for (int idx = tid; idx < NM * NCH; idx += 256) {
        int m = idx >> 4, n = idx & 15;
        td[m * PAD + n] = sd[m * NN + n0 + n];
        tw[m * PAD + n] = sw[m * NN + n0 + n];
    }
    __syncthreads();
    int col = tid & 15, sub = tid >> 4;  // 16 subs x 32 rows each
    for (int r = 0; r < Rr; ++r) {
        float mxd = -3.402823466e38f, mxw = -3.402823466e38f;
        for (int j = 0; j < 32; ++j) {
            int m = sub * 32 + j;
            float g = adj[(r * NM + m) * NN + n0 + col];
            float x = td[m * PAD + col]; if (g == 0.f) x = 0.f; if (x == 0.f) x = NEGV;
            td[m * PAD + col] = x; mxd = fmaxf(mxd, x);
            float y = tw[m * PAD + col]; if (g == 0.f) y = 0.f; if (y == 0.f) y = NEGV;
            tw[m * PAD + col] = y; mxw = fmaxf(mxw, y);
        }
        red[tid] = mxd; __syncthreads();
        float Md = red[col];
        for (int s = 1; s < 16; ++s) Md = fmaxf(Md, red[s * 16 + col]);
        __syncthreads();
        red[tid] = mxw; __syncthreads();
        float Mw = red[col];
        for (int s = 1; s < 16; ++s) Mw = fmaxf(Mw, red[s * 16 + col]);
        __syncthreads();
        float smdl = 0.f, smwl = 0.f;
        for (int j = 0; j < 32; ++j) {
            int m = sub * 32 + j;
            smdl += __expf(td[m * PAD + col] - Md);
            smwl += __expf(tw[m * PAD + col] - Mw);
        }
        red[tid] = smdl; __syncthreads();
        float Sd = 0.f;
        for (int s = 0; s < 16; ++s) Sd += red[s * 16 + col];
        __syncthreads();
        red[tid] = smwl; __syncthreads();
        float Sw = 0.f;
        for (int s = 0; s < 16; ++s) Sw += red[s * 16 + col];
        __syncthreads();
        float rdn = 1.f / Sd, rwn = 1.f / Sw;
        float* po = out + (b * Rr + r) * (NM * NN) + n0 + col;
        for (int j = 0; j < 32; ++j) {
            int m = sub * 32 + j;
            float p1 = __expf(td[m * PAD + col] - Md) * rdn; p1 = (p1 >= DROPV) ? p1 : 0.f;
            float p2 = __expf(tw[m * PAD + col] - Mw) * rwn; p2 = (p2 >= DROPV) ? p2 : 0.f;
            po[m * NN] = 0.5f * (p1 + p2);
        }
        __syncthreads();
    }
}

// ---------------- host launcher ----------------
extern "C" void kernel_launch(void* const* d_in, const int* in_sizes, int n_in,
                              void* d_out, int out_size, void* d_ws, size_t ws_size,
                              hipStream_t stream) {
    const float* xm       = (const float*)d_in[0];
    const float* xn       = (const float*)d_in[1];
    const float* time_x   = (const float*)d_in[2];
    const float* weath_x  = (const float*)d_in[3];
    const float* adj      = (const float*)d_in[4];
    const float* ctw      = (const float*)d_in[5];
    const float* ctb      = (const float*)d_in[6];
    const float* cww      = (const float*)d_in[7];
    const float* cwb      = (const float*)d_in[8];
    const float* cxmw     = (const float*)d_in[9];
    const float* cxmb     = (const float*)d_in[10];
    const float* cxnw     = (const float*)d_in[11];
    const float* cxnb     = (const float*)d_in[12];
    const float* P0       = (const float*)d_in[13];  // m2n_date_P
    const float* P1       = (const float*)d_in[14];  // m2n_weather_P
    const float* P2       = (const float*)d_in[15];  // n2m_date_P
    const float* P3       = (const float*)d_in[16];  // n2m_weather_P
    const float* wmd      = (const float*)d_in[17];
    const float* bmd      = (const float*)d_in[18];
    const float* wmw      = (const float*)d_in[19];
    const float* bmw      = (const float*)d_in[20];
    const float* wnd      = (const float*)d_in[21];
    const float* bnd      = (const float*)d_in[22];
    const float* wnw      = (const float*)d_in[23];
    const float* bnw      = (const float*)d_in[24];

    float* ws  = (float*)d_ws;
    float* zd  = ws + OFF_ZD;
    float* zw  = ws + OFF_ZW;
    float* zc  = ws + OFF_ZC;
    float* ex  = ws + OFF_EX;
    float* Xm  = ws + OFF_XM;
    float* Xn  = ws + OFF_XN;
    float* qk  = ws + OFF_QK;
    float* sim = ws + OFF_SIM;
    float* outp = (float*)d_out;

    conv_z_kernel<<<(Bsz * H4 * Tt + 255) / 256, 256, 0, stream>>>(
        time_x, ctw, ctb, weath_x, cww, cwb, zd, zw);
    build_x_kernel<<<(Bsz * NM * KE + 255) / 256, 256, 0, stream>>>(
        xm, xn, cxmw, cxmb, cxnw, cxnb, Xm, Xn);
    zcontrib_kernel<<<(4 * Bsz * Hh + 255) / 256, 256, 0, stream>>>(
        zd, zw, wmd, wmw, wnd, wnw, zc);
    embed_gemm_kernel<<<dim3(64, 4, 4), 256, 0, stream>>>(
        Xm, Xn, wmd, wmw, wnd, wnw, bmd, bmw, bnd, bnw, zc, ex);
    qk_gemm_kernel<<<dim3(64, 4, 4), 256, 0, stream>>>(ex, P0, P1, P2, P3, qk);
    sim_gemm_kernel<<<dim3(32, 16, 4), 256, 0, stream>>>(qk, ex, sim);
    m2n_softmax_kernel<<<Bsz * NM / 8, 256, 0, stream>>>(sim, adj, outp);
    size_t lds_bytes = (size_t)(2 * NM * PAD + 256) * sizeof(float);  // ~70 KB
    n2m_softmax_kernel<<<dim3(NN / NCH, Bsz), 256, lds_bytes, stream>>>(
        sim, adj, outp + (size_t)Bsz * Rr * NM * NN);
}